// UltraMem_34419867910787
// MI455X (gfx1250) — compile-verified
//
#include <hip/hip_runtime.h>
#include <math.h>

typedef __attribute__((ext_vector_type(2))) float v2f;
typedef __attribute__((ext_vector_type(8))) float v8f;

#define BB     2
#define NSEQ   1024
#define DIMX   1024
#define DQKX   128
#define NKEYS  256
#define TOPKX  32
#define HEADSX 2
#define DVX    512
#define KCONVX 5

#define BN_TOT (BB*NSEQ)           // 2048
#define OUT_ELEMS ((size_t)BB*NSEQ*HEADSX*DVX)  // 2097152

// ---------------- workspace layout (float offsets) ----------------
#define XN_OFF   ((size_t)0)                       // 2048*1024
#define XC_OFF   (XN_OFF + (size_t)BN_TOT*DIMX)    // 2048*1024
#define QRAW_OFF (XC_OFF + (size_t)BN_TOT*DIMX)    // 2048*128
#define Q_OFF    (QRAW_OFF + (size_t)BN_TOT*DQKX)  // 2048*128
#define SC_OFF   (Q_OFF + (size_t)BN_TOT*DQKX)     // 2*2048*256*2
#define UV_OFF   (SC_OFF + (size_t)2*BN_TOT*NKEYS*2)
#define TV_OFF   (UV_OFF + 4)

// ---------------- RMSNorm ----------------
__global__ void rmsnorm_kernel(const float* __restrict__ tokens,
                               const float* __restrict__ rms_w,
                               float* __restrict__ xn) {
  const int row = blockIdx.x;        // bn
  const int tid = threadIdx.x;       // 256
  const float* in = tokens + (size_t)row * DIMX;
  float v[4]; float ss = 0.f;
  #pragma unroll
  for (int i = 0; i < 4; ++i) { v[i] = in[tid + i*256]; ss += v[i]*v[i]; }
  __shared__ float red[256];
  red[tid] = ss; __syncthreads();
  for (int off = 128; off > 0; off >>= 1) {
    if (tid < off) red[tid] += red[tid + off];
    __syncthreads();
  }
  const float scale = rsqrtf(red[0] * (1.0f/DIMX) + 1.1920929e-07f);
  float* o = xn + (size_t)row * DIMX;
  #pragma unroll
  for (int i = 0; i < 4; ++i) { int d = tid + i*256; o[d] = v[i] * scale * rms_w[d]; }
}

// ---------------- causal depthwise conv (K=5, pad left 4) ----------------
__global__ void conv_kernel(const float* __restrict__ xn,
                            const float* __restrict__ conv_w,
                            const float* __restrict__ conv_b,
                            float* __restrict__ xc) {
  const int idx = blockIdx.x * blockDim.x + threadIdx.x;  // over BN_TOT*DIMX
  const int d  = idx & (DIMX - 1);
  const int bn = idx >> 10;
  const int n  = bn & (NSEQ - 1);
  const int b  = bn >> 10;
  float acc = conv_b[d];
  #pragma unroll
  for (int k = 0; k < KCONVX; ++k) {
    const int np = n + k - (KCONVX - 1);
    if (np >= 0)
      acc = fmaf(xn[((size_t)(b*NSEQ + np))*DIMX + d], conv_w[d*KCONVX + k], acc);
  }
  xc[(size_t)idx] = acc;
}

// ---------------- Q = xc @ wq^T   (M=2048,K=1024,N=128) via V_WMMA_F32_16X16X4_F32 ----------------
__global__ void qgemm_kernel(const float* __restrict__ A,   // 2048x1024
                             const float* __restrict__ W,   // 128x1024 (wq)
                             float* __restrict__ Qout) {    // 2048x128
  const int lane  = threadIdx.x;            // wave32
  const int ntile = blockIdx.x;             // 0..7
  const int mtile = blockIdx.y;             // 0..127
  const int mlane = lane & 15;
  const int half  = lane >> 4;
  const float* ap = A + (size_t)(mtile*16 + mlane) * DIMX + 2*half;
  const float* bp = W + (size_t)(ntile*16 + mlane) * DIMX + 2*half;
  v8f acc = {};
  #pragma unroll 4
  for (int k = 0; k < DIMX; k += 4) {
    v2f a = *(const v2f*)(ap + k);
    v2f b = *(const v2f*)(bp + k);
    acc = __builtin_amdgcn_wmma_f32_16x16x4_f32(false, a, false, b, (short)0, acc, false, false);
  }
  #pragma unroll
  for (int i = 0; i < 8; ++i) {
    const int r = mtile*16 + i + 8*half;
    Qout[(size_t)r * DQKX + ntile*16 + mlane] = acc[i];
  }
}

// ---------------- LayerNorm over DQK=128, scale by qln_w ----------------
__global__ void qln_kernel(const float* __restrict__ qraw,
                           const float* __restrict__ qln_w,
                           float* __restrict__ q) {
  const int row = blockIdx.x; const int tid = threadIdx.x; // 128
  __shared__ float red[128];
  const float v = qraw[(size_t)row * DQKX + tid];
  red[tid] = v; __syncthreads();
  for (int off = 64; off > 0; off >>= 1) { if (tid < off) red[tid] += red[tid+off]; __syncthreads(); }
  const float mu = red[0] * (1.0f/DQKX); __syncthreads();
  const float d = v - mu;
  red[tid] = d*d; __syncthreads();
  for (int off = 64; off > 0; off >>= 1) { if (tid < off) red[tid] += red[tid+off]; __syncthreads(); }
  const float var = red[0] * (1.0f/DQKX);
  q[(size_t)row * DQKX + tid] = d * rsqrtf(var + 1e-5f) * qln_w[tid];
}

// ---------------- sc = q @ keys^T  (M=2048,K=128,N=1024) -> SC[c][bn][m][r] ----------------
__global__ void scgemm_kernel(const float* __restrict__ Q,     // 2048x128
                              const float* __restrict__ keys,  // 1024x128 (2,2,256,128)
                              float* __restrict__ SC) {        // [2][2048][256][2]
  const int lane  = threadIdx.x;
  const int ntile = blockIdx.x;   // 0..63
  const int mtile = blockIdx.y;   // 0..127
  const int mlane = lane & 15;
  const int half  = lane >> 4;
  const float* ap = Q    + (size_t)(mtile*16 + mlane) * DQKX + 2*half;
  const float* bp = keys + (size_t)(ntile*16 + mlane) * DQKX + 2*half;
  v8f acc = {};
  #pragma unroll
  for (int k = 0; k < DQKX; k += 4) {
    v2f a = *(const v2f*)(ap + k);
    v2f b = *(const v2f*)(bp + k);
    acc = __builtin_amdgcn_wmma_f32_16x16x4_f32(false, a, false, b, (short)0, acc, false, false);
  }
  const int col = ntile*16 + mlane;
  const int c = col >> 9, r = (col >> 8) & 1, m = col & 255;
  #pragma unroll
  for (int i = 0; i < 8; ++i) {
    const int row = mtile*16 + i + 8*half;
    SC[(((size_t)c * BN_TOT + row) * NKEYS + m) * 2 + r] = acc[i];
  }
}

// ---------------- tiny 2x2 SVDs of core + aux ----------------
__global__ void svd_kernel(const float* __restrict__ core,
                           float* __restrict__ uvec, float* __restrict__ tvec,
                           float* __restrict__ auxout) {
  __shared__ float contrib[HEADSX];
  const int h = threadIdx.x;
  if (h < HEADSX) {
    const float a = core[h*4+0], b = core[h*4+1], c = core[h*4+2], d = core[h*4+3];
    // M^T M = [[p,q],[q,r]]
    const float p = a*a + c*c, qv = a*b + c*d, r = b*b + d*d;
    const float tr = 0.5f*(p + r), df = 0.5f*(p - r);
    const float disc = sqrtf(df*df + qv*qv);
    const float l1 = fmaxf(tr + disc, 0.f);
    const float l2 = fmaxf(tr - disc, 0.f);
    const float s1 = sqrtf(l1), s2 = sqrtf(l2);
    float v0, v1;
    if (fabsf(qv) > 1e-30f) {
      v0 = qv; v1 = l1 - p;
      const float w0 = l1 - r, w1 = qv;
      if (w0*w0 + w1*w1 > v0*v0 + v1*v1) { v0 = w0; v1 = w1; }
    } else { if (p >= r) { v0 = 1.f; v1 = 0.f; } else { v0 = 0.f; v1 = 1.f; } }
    const float inv = rsqrtf(v0*v0 + v1*v1);
    v0 *= inv; v1 *= inv;
    // deterministic sign: larger-magnitude component of v1 positive
    const float big = (fabsf(v0) >= fabsf(v1)) ? v0 : v1;
    if (big < 0.f) { v0 = -v0; v1 = -v1; }
    float u0 = a*v0 + b*v1, u1 = c*v0 + d*v1;
    const float ui = (s1 > 1e-30f) ? (1.f/s1) : 0.f;
    u0 *= ui; u1 *= ui;
    const float un = rsqrtf(fmaxf(u0*u0 + u1*u1, 1e-30f));
    u0 *= un; u1 *= un;
    uvec[h*2+0] = u0; uvec[h*2+1] = u1;
    tvec[h*2+0] = v0; tvec[h*2+1] = v1;
    const float rel = fmaxf(s2 - 0.15f, 0.f);
    contrib[h] = rel * rel;     // mean over RANK-1 = 1 element
  }
  __syncthreads();
  if (threadIdx.x == 0) auxout[0] = 0.1f * (contrib[0] + contrib[1]);
}

// ---------------- per-token routing + gather (one block per (h,b,n)) ----------------
__global__ __launch_bounds__(256)
void topk_gather_kernel(const float* __restrict__ SC,
                        const float* __restrict__ uvec, const float* __restrict__ tvec,
                        const float* __restrict__ core,
                        const float* __restrict__ memories,
                        float* __restrict__ out) {
  const int tid = threadIdx.x;                  // 256
  const int n = blockIdx.x, b = blockIdx.y, h = blockIdx.z;
  const int bn = b * NSEQ + n;

  __shared__ float  s_mrow[NKEYS], s_mcol[NKEYS];
  __shared__ float2 s_rsc[NKEYS],  s_csc[NKEYS];
  __shared__ float  s_scores[TOPKX*TOPKX];
  __shared__ int    s_topr[TOPKX], s_topc[TOPKX];
  __shared__ float  s_redv[256];
  __shared__ int    s_redi[256];
  __shared__ float  s_fs[TOPKX];
  __shared__ int    s_fidx[TOPKX];

  const float u0 = uvec[h*2], u1 = uvec[h*2+1];
  const float t0 = tvec[h*2], t1 = tvec[h*2+1];

  {
    const float2 rsc = *(const float2*)&SC[(((size_t)0 * BN_TOT + bn) * NKEYS + tid) * 2];
    const float2 csc = *(const float2*)&SC[(((size_t)1 * BN_TOT + bn) * NKEYS + tid) * 2];
    s_rsc[tid] = rsc; s_csc[tid] = csc;
    s_mrow[tid] = rsc.x*u0 + rsc.y*u1;
    s_mcol[tid] = csc.x*t0 + csc.y*t1;
  }
  __syncthreads();

  // top-32 of m_row (lowest-index tie-break, matches jax top_k)
  for (int it = 0; it < TOPKX; ++it) {
    s_redv[tid] = s_mrow[tid]; s_redi[tid] = tid; __syncthreads();
    for (int off = 128; off > 0; off >>= 1) {
      if (tid < off) {
        const float v2 = s_redv[tid+off]; const int i2 = s_redi[tid+off];
        if (v2 > s_redv[tid] || (v2 == s_redv[tid] && i2 < s_redi[tid])) { s_redv[tid] = v2; s_redi[tid] = i2; }
      }
      __syncthreads();
    }
    if (tid == 0) { const int bi = s_redi[0]; s_topr[it] = bi; s_mrow[bi] = -__builtin_inff(); }
    __syncthreads();
  }
  // top-32 of m_col
  for (int it = 0; it < TOPKX; ++it) {
    s_redv[tid] = s_mcol[tid]; s_redi[tid] = tid; __syncthreads();
    for (int off = 128; off > 0; off >>= 1) {
      if (tid < off) {
        const float v2 = s_redv[tid+off]; const int i2 = s_redi[tid+off];
        if (v2 > s_redv[tid] || (v2 == s_redv[tid] && i2 < s_redi[tid])) { s_redv[tid] = v2; s_redi[tid] = i2; }
      }
      __syncthreads();
    }
    if (tid == 0) { const int bi = s_redi[0]; s_topc[it] = bi; s_mcol[bi] = -__builtin_inff(); }
    __syncthreads();
  }

  // 32x32 rank-2 bilinear scores
  const float c00 = core[h*4+0], c01 = core[h*4+1], c10 = core[h*4+2], c11 = core[h*4+3];
  #pragma unroll
  for (int qq = 0; qq < 4; ++qq) {
    const int p = tid + qq*256;
    const int i = p >> 5, j = p & 31;
    const float2 fr = s_rsc[s_topr[i]];
    const float2 fc = s_csc[s_topc[j]];
    s_scores[p] = fr.x*(c00*fc.x + c01*fc.y) + fr.y*(c10*fc.x + c11*fc.y);
  }
  __syncthreads();

  // top-32 of 1024 scores -> sigmoid weights + memory indices
  for (int it = 0; it < TOPKX; ++it) {
    float bv = -__builtin_inff(); int bp = 0;
    #pragma unroll
    for (int qq = 0; qq < 4; ++qq) {
      const int p = tid + qq*256;
      const float v = s_scores[p];
      if (v > bv) { bv = v; bp = p; }     // ascending p scan: lowest p kept on tie
    }
    s_redv[tid] = bv; s_redi[tid] = bp; __syncthreads();
    for (int off = 128; off > 0; off >>= 1) {
      if (tid < off) {
        const float v2 = s_redv[tid+off]; const int i2 = s_redi[tid+off];
        if (v2 > s_redv[tid] || (v2 == s_redv[tid] && i2 < s_redi[tid])) { s_redv[tid] = v2; s_redi[tid] = i2; }
      }
      __syncthreads();
    }
    if (tid == 0) {
      const int p = s_redi[0]; const float v = s_redv[0];
      s_fs[it]   = 1.f / (1.f + expf(-v));
      s_fidx[it] = s_topr[p >> 5] * NKEYS + s_topc[p & 31];
      s_scores[p] = -__builtin_inff();
    }
    __syncthreads();
  }

  // weighted gather-sum of 32 memory rows (512 f32 each); dominant bandwidth phase
  float acc0 = 0.f, acc1 = 0.f;
  for (int k = 0; k < TOPKX; ++k) {
    const float* mr = memories + (size_t)s_fidx[k] * DVX;
    if (k + 1 < TOPKX) __builtin_prefetch(memories + (size_t)s_fidx[k+1] * DVX, 0, 0);
    const float w = s_fs[k];
    acc0 = fmaf(mr[tid],       w, acc0);
    acc1 = fmaf(mr[tid + 256], w, acc1);
  }
  float* orow = out + ((size_t)bn * HEADSX + h) * DVX;  // out[b][n][h*DV + d]
  orow[tid]       = acc0;
  orow[tid + 256] = acc1;
}

// ---------------- launch ----------------
extern "C" void kernel_launch(void* const* d_in, const int* in_sizes, int n_in,
                              void* d_out, int out_size, void* d_ws, size_t ws_size,
                              hipStream_t stream) {
  (void)in_sizes; (void)n_in; (void)out_size; (void)ws_size;
  const float* tokens   = (const float*)d_in[0];
  const float* rms_w    = (const float*)d_in[1];
  const float* conv_w   = (const float*)d_in[2];
  const float* conv_b   = (const float*)d_in[3];
  const float* wq       = (const float*)d_in[4];
  const float* qln_w    = (const float*)d_in[5];
  const float* keys     = (const float*)d_in[6];
  const float* core     = (const float*)d_in[7];
  const float* memories = (const float*)d_in[8];

  float* ws   = (float*)d_ws;
  float* xn   = ws + XN_OFF;
  float* xc   = ws + XC_OFF;
  float* qraw = ws + QRAW_OFF;
  float* q    = ws + Q_OFF;
  float* sc   = ws + SC_OFF;
  float* uv   = ws + UV_OFF;
  float* tv   = ws + TV_OFF;

  float* out  = (float*)d_out;
  float* aux  = out + OUT_ELEMS;

  rmsnorm_kernel<<<BN_TOT, 256, 0, stream>>>(tokens, rms_w, xn);
  conv_kernel<<<(BN_TOT*DIMX)/256, 256, 0, stream>>>(xn, conv_w, conv_b, xc);
  qgemm_kernel<<<dim3(DQKX/16, BN_TOT/16), 32, 0, stream>>>(xc, wq, qraw);
  qln_kernel<<<BN_TOT, DQKX, 0, stream>>>(qraw, qln_w, q);
  scgemm_kernel<<<dim3(1024/16, BN_TOT/16), 32, 0, stream>>>(q, keys, sc);
  svd_kernel<<<1, 32, 0, stream>>>(core, uv, tv, aux);
  topk_gather_kernel<<<dim3(NSEQ, BB, HEADSX), 256, 0, stream>>>(sc, uv, tv, core, memories, out);
}